// MultiHeadCrossAttentionFusion_6227702579448
// MI455X (gfx1250) — compile-verified
//
#include <hip/hip_runtime.h>

typedef __attribute__((ext_vector_type(2))) float v2f;
typedef __attribute__((ext_vector_type(4))) float v4f;
typedef __attribute__((ext_vector_type(8))) float v8f;

#define HDIM 256   // hidden dim H
#define KIN  512   // TD == ID
#define BM   64    // block M tile
#define KC   32    // K chunk staged in LDS
#define LDSS 34    // padded LDS row stride (floats); keeps b64 operand reads bank-conflict-free

// ---------------------------------------------------------------------------
// Tiny precompute kernels (negligible work: folds the whole network into
// P_img[256,512], P_text[256,512], bias_total[256]).
// ---------------------------------------------------------------------------

// C[m,n] = sum_k A[m*lda + k] * B[k*ldb + n]   (C row-major, stride N)
__global__ void gemm_naive(float* __restrict__ C,
                           const float* __restrict__ A, int lda,
                           const float* __restrict__ B, int ldb,
                           int M, int N, int K) {
    int idx = blockIdx.x * blockDim.x + threadIdx.x;
    if (idx >= M * N) return;
    int m = idx / N, n = idx % N;
    float s = 0.f;
    for (int k = 0; k < K; ++k) s += A[m * lda + k] * B[k * ldb + n];
    C[idx] = s;
}

// c[i] = out_proj_w[i,:] . bv + out_proj_b[i]   (bv = in_proj_b[2H:3H])
__global__ void cvec_kernel(float* __restrict__ c,
                            const float* __restrict__ out_proj_w,
                            const float* __restrict__ in_proj_b,
                            const float* __restrict__ out_proj_b) {
    int i = threadIdx.x;  // 256 threads
    const float* bv = in_proj_b + 2 * HDIM;
    float s = out_proj_b[i];
    for (int k = 0; k < HDIM; ++k) s += out_proj_w[i * HDIM + k] * bv[k];
    c[i] = s;
}

// bias_total[n] = G_img[n,:].b_img + G_txt[n,:].b_text
//               + W_fuse[n,0:256].c + W_fuse[n,256:512].c + b_fuse[n]
__global__ void bias_kernel(float* __restrict__ bias,
                            const float* __restrict__ Gimg,
                            const float* __restrict__ Gtxt,
                            const float* __restrict__ b_img,
                            const float* __restrict__ b_text,
                            const float* __restrict__ W_fuse,
                            const float* __restrict__ cvec,
                            const float* __restrict__ b_fuse) {
    int n = threadIdx.x;  // 256 threads
    float s = b_fuse[n];
    for (int k = 0; k < HDIM; ++k) s += Gimg[n * HDIM + k] * b_img[k];
    for (int k = 0; k < HDIM; ++k) s += Gtxt[n * HDIM + k] * b_text[k];
    for (int k = 0; k < HDIM; ++k) s += W_fuse[n * 2 * HDIM + k] * cvec[k];
    for (int k = 0; k < HDIM; ++k) s += W_fuse[n * 2 * HDIM + HDIM + k] * cvec[k];
    bias[n] = s;
}

// ---------------------------------------------------------------------------
// Main fused kernel:
//   out[m,n] = relu( sum_k Ximg[m,k]*Pimg[n,k] + sum_k Xtxt[m,k]*Ptxt[n,k] + bias[n] )
// Block: 256 threads (8 wave32s), tile 64(M) x 256(N), K staged in LDS (KC=32).
// Wave (wm, wn): wm in 0..3 picks the 16-row M tile, wn in 0..1 picks the
// 128-col N half; each wave runs 8 WMMAs per K-step sharing one A fragment.
// ---------------------------------------------------------------------------
__global__ __launch_bounds__(256) void fused_dual_gemm_relu(
    const float* __restrict__ Ximg, const float* __restrict__ Xtxt,
    const float* __restrict__ Pimg, const float* __restrict__ Ptxt,
    const float* __restrict__ bias, float* __restrict__ out) {

    __shared__ float sA[BM * LDSS];     //  8,704 B
    __shared__ float sB[HDIM * LDSS];   // 34,816 B

    const int tid   = threadIdx.x;
    const int lane  = tid & 31;
    const int wave  = tid >> 5;   // 0..7
    const int wm    = wave & 3;   // M tile index
    const int wn    = wave >> 2;  // N half (0/1)
    const int l16   = lane & 15;
    const int lhalf = lane >> 4;  // 0/1 -> K offset +2 per ISA A/B layout
    const int m0    = blockIdx.x * BM;

    // accumulators: 8 N-tiles of 16x16, v8f each (ISA C/D layout)
    v8f acc[8];
#pragma unroll
    for (int t = 0; t < 8; ++t)
#pragma unroll
        for (int r = 0; r < 8; ++r) acc[t][r] = 0.0f;

    // staging coordinates: thread t -> row t/4, 8 floats at col (t%4)*8
    const int srow = tid >> 2;
    const int scol = (tid & 3) * 8;

    for (int phase = 0; phase < 2; ++phase) {
        const float* __restrict__ X = phase ? Xtxt : Ximg;
        const float* __restrict__ P = phase ? Ptxt : Pimg;

        for (int kb = 0; kb < KIN; kb += KC) {
            __syncthreads();
            // ---- stage A chunk: 64 x 32 (coalesced b128 loads) ----
            {
                const float* g = X + (size_t)(m0 + srow) * KIN + kb + scol;
                v4f a0 = *(const v4f*)(g);
                v4f a1 = *(const v4f*)(g + 4);
                float* s = sA + srow * LDSS + scol;
                *(v4f*)(s)     = a0;
                *(v4f*)(s + 4) = a1;
            }
            // ---- stage B chunk: 256 x 32, four passes of 64 rows ----
#pragma unroll
            for (int p = 0; p < 4; ++p) {
                int r = p * 64 + srow;
                const float* g = P + (size_t)r * KIN + kb + scol;
                v4f b0 = *(const v4f*)(g);
                v4f b1 = *(const v4f*)(g + 4);
                float* s = sB + r * LDSS + scol;
                *(v4f*)(s)     = b0;
                *(v4f*)(s + 4) = b1;
            }
            __syncthreads();

            // ---- compute: 8 K-steps of 4 (v_wmma_f32_16x16x4_f32) ----
#pragma unroll
            for (int ks = 0; ks < KC; ks += 4) {
                // A frag: lanes 0-15 -> A[m, ks..ks+1]; lanes 16-31 -> A[m, ks+2..ks+3]
                v2f afrag = *(const v2f*)(sA + (wm * 16 + l16) * LDSS + ks + 2 * lhalf);
#pragma unroll
                for (int t = 0; t < 8; ++t) {
                    int n = wn * 128 + t * 16 + l16;
                    // B frag: B[k',n'] = P[n, k]; same lane-striped b64 pattern
                    v2f bfrag = *(const v2f*)(sB + n * LDSS + ks + 2 * lhalf);
                    acc[t] = __builtin_amdgcn_wmma_f32_16x16x4_f32(
                        /*neg_a=*/false, afrag, /*neg_b=*/false, bfrag,
                        /*c_mod=*/(short)0, acc[t],
                        /*reuse_a=*/false, /*reuse_b=*/false);
                }
            }
        }
    }

    // ---- epilogue: bias + relu, D layout VGPR r -> M = r + 8*lhalf ----
    const int mbase = m0 + wm * 16 + 8 * lhalf;
#pragma unroll
    for (int t = 0; t < 8; ++t) {
        int n = wn * 128 + t * 16 + l16;
        float b = bias[n];
#pragma unroll
        for (int r = 0; r < 8; ++r) {
            float v = acc[t][r] + b;
            v = v > 0.0f ? v : 0.0f;
            out[(size_t)(mbase + r) * HDIM + n] = v;
        }
    }
}

// ---------------------------------------------------------------------------
extern "C" void kernel_launch(void* const* d_in, const int* in_sizes, int n_in,
                              void* d_out, int out_size, void* d_ws, size_t ws_size,
                              hipStream_t stream) {
    const float* text_features  = (const float*)d_in[0];
    const float* image_features = (const float*)d_in[1];
    const float* W_text     = (const float*)d_in[2];
    const float* b_text     = (const float*)d_in[3];
    const float* W_img      = (const float*)d_in[4];
    const float* b_img      = (const float*)d_in[5];
    const float* in_proj_w  = (const float*)d_in[6];
    const float* in_proj_b  = (const float*)d_in[7];
    const float* out_proj_w = (const float*)d_in[8];
    const float* out_proj_b = (const float*)d_in[9];
    const float* W_fuse     = (const float*)d_in[10];
    const float* b_fuse     = (const float*)d_in[11];
    float* out = (float*)d_out;
    float* ws  = (float*)d_ws;

    // workspace layout (floats): ~1.84 MB total
    float* Mmat = ws;            // 256x256
    float* Gimg = ws + 65536;    // 256x256
    float* Gtxt = ws + 131072;   // 256x256
    float* Pimg = ws + 196608;   // 256x512
    float* Ptxt = ws + 327680;   // 256x512
    float* bias = ws + 458752;   // 256
    float* cvec = ws + 459008;   // 256

    const float* Wv = in_proj_w + 2 * HDIM * HDIM;  // rows 2H..3H of in_proj_w

    // fold the network: M = out_proj_w @ Wv ; G = W_fuse_half @ M ; P = G @ W_in
    cvec_kernel<<<1, 256, 0, stream>>>(cvec, out_proj_w, in_proj_b, out_proj_b);
    gemm_naive<<<256, 256, 0, stream>>>(Mmat, out_proj_w, HDIM, Wv, HDIM, HDIM, HDIM, HDIM);
    gemm_naive<<<256, 256, 0, stream>>>(Gimg, W_fuse, 2 * HDIM, Mmat, HDIM, HDIM, HDIM, HDIM);
    gemm_naive<<<256, 256, 0, stream>>>(Gtxt, W_fuse + HDIM, 2 * HDIM, Mmat, HDIM, HDIM, HDIM, HDIM);
    gemm_naive<<<512, 256, 0, stream>>>(Pimg, Gimg, HDIM, W_img, KIN, HDIM, KIN, HDIM);
    gemm_naive<<<512, 256, 0, stream>>>(Ptxt, Gtxt, HDIM, W_text, KIN, HDIM, KIN, HDIM);
    bias_kernel<<<1, 256, 0, stream>>>(bias, Gimg, Gtxt, b_img, b_text, W_fuse, cvec, b_fuse);

    // main fused dual-GEMM + bias + relu (memory-bound: ~335 MB @ 23.3 TB/s)
    const int B = in_sizes[0] / KIN;  // 65536
    fused_dual_gemm_relu<<<B / BM, 256, 0, stream>>>(
        image_features, text_features, Pimg, Ptxt, bias, out);
}